// ModernBertTokenCRF_43026982371981
// MI455X (gfx1250) — compile-verified
//
#include <hip/hip_runtime.h>
#include <hip/hip_bf16.h>

// CRF loss for MI455X (gfx1250).
// Forward algorithm in scaled-linear space so the per-step log-semiring
// vecmat becomes a real matmul done with v_wmma_f32_16x16x4_f32.
// Emissions tiles staged LDS via Tensor Data Mover (tensor_load_to_lds),
// double buffered with s_wait_tensorcnt.

#define BB 512
#define TT 2048
#define LL 37   // number of states
#define KP 40   // K padded to multiple of 4
#define PSTR 40 // row stride of P in LDS (floats)

typedef __attribute__((ext_vector_type(2))) float v2f;
typedef __attribute__((ext_vector_type(8))) float v8f;
typedef __attribute__((ext_vector_type(4))) unsigned int v4u;
typedef __attribute__((ext_vector_type(8))) int v8i;
typedef __attribute__((ext_vector_type(4))) int v4i;

// ---------------------------------------------------------------------------
// TDM: load a 2D tile (16 rows x LL floats, row stride TT*LL floats) into LDS.
// Descriptor per CDNA5 ISA (groups 0/1; groups 2/3 zero).
// This toolchain exposes the 6-arg builtin:
//   (uint32x4 g0, int32x8 g1, int32x4 g2, int32x4 g3, int32x8 g4, i32 cpol)
// ---------------------------------------------------------------------------
__device__ __forceinline__ void tdm_load_tile(unsigned lds_off, const float* gaddr) {
    unsigned long long ga = (unsigned long long)(size_t)gaddr;
    v4u g0;
    g0[0] = 1u;                                   // count=1, user descriptor
    g0[1] = lds_off;                              // lds_addr (bytes)
    g0[2] = (unsigned)(ga & 0xffffffffu);         // global_addr[31:0]
    g0[3] = (unsigned)((ga >> 32) & 0x01ffffffu)  // global_addr[56:32]
            | (2u << 30);                         // type = 2 ("image")
    unsigned long long s0 = (unsigned long long)TT * (unsigned long long)LL; // dim0 stride
    v8i g1;
    g1[0] = (int)(2u << 16);                      // data_size = 4B; no multicast/pad
    g1[1] = (int)(((unsigned)LL & 0xffffu) << 16);       // tensor_dim0 lo16 @ bit48
    g1[2] = (int)((((unsigned)LL >> 16) & 0xffffu)       // tensor_dim0 hi16
            | ((16u & 0xffffu) << 16));                   // tensor_dim1 lo16
    g1[3] = (int)(((16u >> 16) & 0xffffu)                 // tensor_dim1 hi16
            | (((unsigned)LL & 0xffffu) << 16));          // tile_dim0 = 37
    g1[4] = (int)(16u);                            // tile_dim1 = 16, tile_dim2 = 0
    g1[5] = (int)(unsigned)(s0 & 0xffffffffu);     // tensor_dim0_stride lo32
    g1[6] = (int)(unsigned)((s0 >> 32) & 0xffffu); // stride0 hi16 | stride1 lo16 (0)
    g1[7] = 0;                                     // stride1 hi32
    v4i z4 = {0, 0, 0, 0};
    v8i z8 = {0, 0, 0, 0, 0, 0, 0, 0};
    __builtin_amdgcn_tensor_load_to_lds(g0, g1, z4, z4, z8, 0);
}

// ---------------------------------------------------------------------------
// Forward-scan kernel: one wave (32 lanes) per 16-batch tile; grid = B/16.
// State per wave: P (16xKP normalized probs) held in WMMA A-fragments,
// logZ per row. B-fragments = exp(transitions), loop invariant in VGPRs.
// ---------------------------------------------------------------------------
__global__ __launch_bounds__(32) void crf_scan_kernel(
    const float* __restrict__ emissions,
    const float* __restrict__ transitions,
    const float* __restrict__ start_t,
    const float* __restrict__ end_t,
    const int* __restrict__ mask,
    float* __restrict__ ws)
{
    __shared__ float EM[2][16 * LL];  // raw emissions tiles (double buffered, TDM dest)
    __shared__ float P[16 * PSTR];    // normalized probabilities, padded cols = 0
    __shared__ float RM[16];          // per-row max (scale) of current step
    __shared__ float EE[KP];          // exp(end_transitions), padded 0

    const int l  = threadIdx.x;
    const int hi = l >> 4;   // 0 or 1
    const int lo = l & 15;
    const int b0 = blockIdx.x * 16;

    // ---- loop-invariant B fragments: exp(transitions), zero padded ----
    v2f Bf[10][3];
#pragma unroll
    for (int kt = 0; kt < 10; ++kt) {
#pragma unroll
        for (int nt = 0; nt < 3; ++nt) {
            int k0 = kt * 4 + 2 * hi;
            int n  = nt * 16 + lo;
            float x0 = (k0     < LL && n < LL) ? __expf(transitions[k0 * LL + n])       : 0.0f;
            float x1 = (k0 + 1 < LL && n < LL) ? __expf(transitions[(k0 + 1) * LL + n]) : 0.0f;
            Bf[kt][nt].x = x0;
            Bf[kt][nt].y = x1;
        }
    }

    // ---- exp(end), zero fill of P padding ----
    for (int k = l; k < KP; k += 32) EE[k] = (k < LL) ? __expf(end_t[k]) : 0.0f;
    for (int i = l; i < 16 * PSTR; i += 32) P[i] = 0.0f;
    __syncthreads();

    // ---- alpha0 = start + emissions[:,0,:]  (scaled-linear init) ----
    float logZ = 0.0f;
    if (l < 16) {
        const int r = l;
        const float* em0 = emissions + (size_t)(b0 + r) * TT * LL;  // t = 0
        float m = -3.4e38f;
        for (int k = 0; k < LL; ++k) m = fmaxf(m, start_t[k] + em0[k]);
        for (int k = 0; k < LL; ++k) P[r * PSTR + k] = __expf(start_t[k] + em0[k] - m);
        logZ = m;
    }
    logZ = __shfl(logZ, lo);  // replicate to upper half (same row)
    __syncthreads();

    // ---- load A fragments from P ----
    v2f Af[10];
#pragma unroll
    for (int kt = 0; kt < 10; ++kt) {
        int k = kt * 4 + 2 * hi;  // even -> 8B aligned
        Af[kt] = *(const v2f*)&P[lo * PSTR + k];
    }

    // LDS byte offsets of the two emission buffers (flat addr truncation;
    // LDS aperture places the offset in the low 32 bits).
    const unsigned em_off0 = (unsigned)(size_t)&EM[0][0];
    const unsigned em_off1 = (unsigned)(size_t)&EM[1][0];

    // prefetch t = 1
    tdm_load_tile(em_off1, emissions + ((size_t)b0 * TT + 1) * LL);

    for (int t = 1; t < TT; ++t) {
        // issue next tile, then wait for the current one
        if (t + 1 < TT) {
            tdm_load_tile(((t + 1) & 1) ? em_off1 : em_off0,
                          emissions + ((size_t)b0 * TT + (t + 1)) * LL);
            __builtin_amdgcn_s_wait_tensorcnt(1);
        } else {
            __builtin_amdgcn_s_wait_tensorcnt(0);
        }

        const int mk = mask[(size_t)(b0 + lo) * TT + t];  // row lo's mask at t

        // ---- Q = P @ exp(T) via WMMA f32 16x16x4, K = 40 ----
        v8f acc0 = {}; v8f acc1 = {}; v8f acc2 = {};
#pragma unroll
        for (int kt = 0; kt < 10; ++kt) {
            acc0 = __builtin_amdgcn_wmma_f32_16x16x4_f32(false, Af[kt], false, Bf[kt][0],
                                                         (short)0, acc0, false, false);
            acc1 = __builtin_amdgcn_wmma_f32_16x16x4_f32(false, Af[kt], false, Bf[kt][1],
                                                         (short)0, acc1, false, false);
            acc2 = __builtin_amdgcn_wmma_f32_16x16x4_f32(false, Af[kt], false, Bf[kt][2],
                                                         (short)0, acc2, false, false);
        }

        // ---- multiply by exp(emissions[:, t, :]); row max ----
        const float* EMt = EM[t & 1];
        float rm[8];
#pragma unroll
        for (int v = 0; v < 8; ++v) {
            int m = v + 8 * hi;           // batch row within tile
            float q0 = acc0[v] * __expf(EMt[m * LL + lo]);        // n = lo
            float q1 = acc1[v] * __expf(EMt[m * LL + 16 + lo]);   // n = 16+lo (< 32)
            float q2 = 0.0f;
            int n2 = 32 + lo;
            if (n2 < LL) q2 = acc2[v] * __expf(EMt[m * LL + n2]);
            acc0[v] = q0; acc1[v] = q1; acc2[v] = q2;
            rm[v] = fmaxf(fmaxf(q0, q1), q2);
        }
        // reduce row max across the 16 lanes of each half
#pragma unroll
        for (int off = 1; off < 16; off <<= 1) {
#pragma unroll
            for (int v = 0; v < 8; ++v) rm[v] = fmaxf(rm[v], __shfl_xor(rm[v], off));
        }

        // ---- normalize & stage P for next step ----
#pragma unroll
        for (int v = 0; v < 8; ++v) {
            int m = v + 8 * hi;
            float riv = 1.0f / rm[v];
            P[m * PSTR + lo]      = acc0[v] * riv;
            P[m * PSTR + 16 + lo] = acc1[v] * riv;
            if (32 + lo < LL) P[m * PSTR + 32 + lo] = acc2[v] * riv;
            if (lo == 0) RM[m] = rm[v];
        }
        __syncthreads();

        // ---- update A fragments + logZ (masked rows keep old state) ----
        const bool keep = (mk != 0);
        const float lrm = __logf(RM[lo]);
#pragma unroll
        for (int kt = 0; kt < 10; ++kt) {
            int k = kt * 4 + 2 * hi;
            v2f nA = *(const v2f*)&P[lo * PSTR + k];
            Af[kt].x = keep ? nA.x : Af[kt].x;
            Af[kt].y = keep ? nA.y : Af[kt].y;
        }
        logZ += keep ? lrm : 0.0f;
        __syncthreads();  // protect P/RM against next iteration's writes
    }

    // ---- denominator: logZ + log(sum_k P[r,k] * exp(end[k])) ----
    float s = 0.0f;
#pragma unroll
    for (int kt = 0; kt < 10; ++kt) {
        int k = kt * 4 + 2 * hi;
        s += Af[kt].x * EE[k] + Af[kt].y * EE[k + 1];
    }
    s += __shfl_xor(s, 16);  // combine K-halves of the same row
    if (l < 16) {
        float denom = logZ + __logf(s);
        atomicAdd(&ws[1], denom);
    }
}

// ---------------------------------------------------------------------------
// Numerator: gold-path score, one 256-thread block per batch row.
// ---------------------------------------------------------------------------
__global__ __launch_bounds__(256) void crf_num_kernel(
    const float* __restrict__ emissions,
    const float* __restrict__ transitions,
    const float* __restrict__ start_t,
    const float* __restrict__ end_t,
    const int* __restrict__ labels,
    const int* __restrict__ mask,
    float* __restrict__ ws)
{
    const int b = blockIdx.x;
    const int tid = threadIdx.x;
    const float* emb = emissions + (size_t)b * TT * LL;
    const int* lab = labels + (size_t)b * TT;
    const int* mk  = mask + (size_t)b * TT;

    float s = 0.0f;
    int cnt = 0;
    for (int t = tid; t < TT; t += 256) {
        int m = mk[t];
        cnt += (m != 0);
        if (t >= 1 && m != 0) {
            int lp = lab[t - 1], lc = lab[t];
            s += transitions[lp * LL + lc] + emb[(size_t)t * LL + lc];
        }
    }
    __shared__ float rs[256];
    __shared__ int   rc[256];
    rs[tid] = s; rc[tid] = cnt;
    __syncthreads();
    for (int off = 128; off > 0; off >>= 1) {
        if (tid < off) { rs[tid] += rs[tid + off]; rc[tid] += rc[tid + off]; }
        __syncthreads();
    }
    if (tid == 0) {
        int len = rc[0];
        int l0 = lab[0];
        int llast = lab[(len > 0 ? len : 1) - 1];
        float tot = rs[0] + start_t[l0] + emb[l0] + end_t[llast];
        atomicAdd(&ws[0], tot);
        atomicAdd(&ws[2], (float)len);
    }
}

__global__ void crf_init(float* ws) {
    if (threadIdx.x < 3) ws[threadIdx.x] = 0.0f;
}

__global__ void crf_finalize(const float* __restrict__ ws, float* __restrict__ out) {
    if (threadIdx.x == 0 && blockIdx.x == 0) {
        float ntok = fmaxf(ws[2], 1.0f);
        out[0] = -(ws[0] - ws[1]) / ntok;
    }
}

extern "C" void kernel_launch(void* const* d_in, const int* in_sizes, int n_in,
                              void* d_out, int out_size, void* d_ws, size_t ws_size,
                              hipStream_t stream) {
    (void)in_sizes; (void)n_in; (void)out_size; (void)ws_size;
    const float* emissions   = (const float*)d_in[0];
    const float* transitions = (const float*)d_in[1];
    const float* start_t     = (const float*)d_in[2];
    const float* end_t       = (const float*)d_in[3];
    const int*   labels      = (const int*)d_in[4];
    const int*   amask       = (const int*)d_in[5];
    float* out = (float*)d_out;
    float* ws  = (float*)d_ws;

    crf_init<<<1, 32, 0, stream>>>(ws);
    crf_num_kernel<<<BB, 256, 0, stream>>>(emissions, transitions, start_t, end_t,
                                           labels, amask, ws);
    crf_scan_kernel<<<BB / 16, 32, 0, stream>>>(emissions, transitions, start_t,
                                                end_t, amask, ws);
    crf_finalize<<<1, 32, 0, stream>>>(ws, out);
}